// LowRank2d_40664750359240
// MI455X (gfx1250) — compile-verified
//
#include <hip/hip_runtime.h>

typedef __attribute__((ext_vector_type(2))) float v2f;
typedef __attribute__((ext_vector_type(8))) float v8f;

#define NX    16384
#define HID   128
#define ROWS  256    // B*CI == B*CO

__device__ __forceinline__ v8f wmma4(v2f a, v2f b, v8f c) {
  // V_WMMA_F32_16X16X4_F32 : D = A(16x4 f32) x B(4x16 f32) + C(16x16 f32)
  return __builtin_amdgcn_wmma_f32_16x16x4_f32(
      /*neg_a=*/false, a, /*neg_b=*/false, b,
      /*c_mod=*/(short)0, c, /*reuse_a=*/false, /*reuse_b=*/false);
}

// ---------------- Kernel 1: MLP trunks (2->64->128, relu) ----------------
// H2P[c][n] = h2_phi feature-major (B-matrix of GEMM1, K=n contiguous)
// H2S[n][c] = h2_psi point-major   (B-matrix of GEMM2, K=c contiguous)
__global__ void mlp_trunk(const float* __restrict__ pw1, const float* __restrict__ pb1,
                          const float* __restrict__ pw2, const float* __restrict__ pb2,
                          const float* __restrict__ qw1, const float* __restrict__ qb1,
                          const float* __restrict__ qw2, const float* __restrict__ qb2,
                          float* __restrict__ H2P, float* __restrict__ H2S) {
  __shared__ float s_w1[128];
  __shared__ float s_b1[64];
  __shared__ float s_w2[64 * 128];
  __shared__ float s_b2[128];
  const int tid = threadIdx.x;                 // 0..127
  const int n = blockIdx.x * 128 + tid;        // grid point
  const float gx = (float)(n >> 7) * (1.0f / 127.0f);
  const float gy = (float)(n & 127) * (1.0f / 127.0f);

  for (int net = 0; net < 2; ++net) {
    const float* w1 = net ? qw1 : pw1;
    const float* b1 = net ? qb1 : pb1;
    const float* w2 = net ? qw2 : pw2;
    const float* b2 = net ? qb2 : pb2;
    __syncthreads();
    s_w1[tid] = w1[tid];
    if (tid < 64) s_b1[tid] = b1[tid];
    for (int i = tid; i < 64 * 128; i += 128) s_w2[i] = w2[i];
    s_b2[tid] = b2[tid];
    __syncthreads();

    float h1[64];
#pragma unroll
    for (int j = 0; j < 64; ++j) {
      float v = gx * s_w1[j] + gy * s_w1[64 + j] + s_b1[j];
      h1[j] = fmaxf(v, 0.0f);
    }
    for (int c = 0; c < HID; ++c) {
      float acc = s_b2[c];
#pragma unroll 16
      for (int j = 0; j < 64; ++j) acc += h1[j] * s_w2[j * 128 + c];
      acc = fmaxf(acc, 0.0f);
      if (net == 0) H2P[c * NX + n] = acc;
      else          H2S[n * HID + c] = acc;
    }
  }
}

// ---------------- Kernel 2: X1[r] = sum_n x[r][n]  (r = b*CI+i) --------------
__global__ void rowsum(const float* __restrict__ x, float* __restrict__ X1) {
  __shared__ float red[256];
  const int r = blockIdx.x;
  float s = 0.0f;
  for (int n = threadIdx.x; n < NX; n += 256) s += x[(long)r * NX + n];
  red[threadIdx.x] = s;
  __syncthreads();
  for (int off = 128; off > 0; off >>= 1) {
    if ((int)threadIdx.x < off) red[threadIdx.x] += red[threadIdx.x + off];
    __syncthreads();
  }
  if (threadIdx.x == 0) X1[r] = red[0];
}

// ---------------- Kernel 3: GEMM1 partials over K-chunks ---------------------
// P[kc][r][c] = sum_{n in chunk kc} x[r][n] * H2P[c][n]
// wave-per-(Mstrip, Kchunk); one wave computes a 16x128 strip.
// Pointer-bump addressing; prefetch the streaming x row 256 B ahead.
__global__ void gemm1_partial(const float* __restrict__ X, const float* __restrict__ H2P,
                              float* __restrict__ P) {
  const int lane = threadIdx.x;        // 0..31
  const int half = lane >> 4;          // 0/1
  const int l16  = lane & 15;
  const int ms   = blockIdx.x;         // 0..15 : M strip of 16 rows
  const int kc   = blockIdx.y;         // 0..63 : K chunk of 256
  const long k0  = (long)kc * 256 + 2 * half;   // lane's K pair start

  v8f acc[8];
  const v8f vzero = {0.f,0.f,0.f,0.f,0.f,0.f,0.f,0.f};
#pragma unroll
  for (int t = 0; t < 8; ++t) acc[t] = vzero;

  const float* aptr = X   + (long)(ms * 16 + l16) * NX + k0;  // A[row][K..K+1]
  const float* bptr = H2P + (long)l16 * NX + k0;              // tile t: +t*16*NX
  for (int kb = 0; kb < 256; kb += 4) {
    __builtin_prefetch(aptr + 64, 0, 0);        // stream x: global_prefetch_b8
    v2f a = *(const v2f*)aptr;
#pragma unroll
    for (int t = 0; t < 8; ++t) {
      v2f b = *(const v2f*)(bptr + (long)t * 16 * NX);  // B[K][t*16+l16] pair
      acc[t] = wmma4(a, b, acc[t]);
    }
    aptr += 4;
    bptr += 4;
  }
#pragma unroll
  for (int t = 0; t < 8; ++t)
#pragma unroll
    for (int r = 0; r < 8; ++r)
      P[((long)kc * ROWS + ms * 16 + r + 8 * half) * HID + t * 16 + l16] = acc[t][r];
}

// ---------------- Kernel 4: deterministic serial reduce of partials ------------
__global__ void reduceP(const float* __restrict__ P, float* __restrict__ T) {
  const int idx = blockIdx.x * 256 + threadIdx.x;   // 0..32767 = ROWS*HID
  float s = 0.0f;
  for (int kc = 0; kc < 64; ++kc) s += P[(long)kc * (ROWS * HID) + idx];
  T[idx] = s;
}

// ---------------- Kernel 5: s[b,o,i] = T[r]·w3p[:,oi] + b3p[oi]·X1[r] --------
__global__ void s_kernel(const float* __restrict__ T, const float* __restrict__ w3p,
                         const float* __restrict__ b3p, const float* __restrict__ X1,
                         float* __restrict__ S) {
  const int idx = blockIdx.x * 256 + threadIdx.x;   // 0..8191
  const int b = idx >> 10;
  const int o = (idx >> 5) & 31;
  const int i = idx & 31;
  const int r = b * 32 + i;
  const int oi = o * 32 + i;
  float acc = b3p[oi] * X1[r];
  for (int k = 0; k < HID; ++k) acc += T[r * HID + k] * w3p[k * 1024 + oi];
  S[idx] = acc;                                     // S[b][o][i]
}

// ---------------- Kernel 6: U[bo][k] = sum_i S[bo][i]*w3q[k][o*32+i] ---------
__global__ void u_kernel(const float* __restrict__ S, const float* __restrict__ w3q,
                         float* __restrict__ U) {
  const int idx = blockIdx.x * 256 + threadIdx.x;   // 0..32767
  const int bo = idx >> 7;
  const int k = idx & 127;
  const int o = bo & 31;
  float acc = 0.0f;
#pragma unroll
  for (int i = 0; i < 32; ++i) acc += S[bo * 32 + i] * w3q[k * 1024 + o * 32 + i];
  U[idx] = acc;
}

// ---------------- Kernel 7: c[bo] = sum_i S[bo][i]*b3q[o*32+i] ---------------
__global__ void c_kernel(const float* __restrict__ S, const float* __restrict__ b3q,
                         float* __restrict__ C) {
  const int bo = threadIdx.x;                       // 0..255
  const int o = bo & 31;
  float acc = 0.0f;
#pragma unroll
  for (int i = 0; i < 32; ++i) acc += S[bo * 32 + i] * b3q[o * 32 + i];
  C[bo] = acc;
}

// ---------------- Kernel 8: out[bo][m] = (U[bo]·H2S[m] + c[bo]) / Nx ---------
// wave per 16x64 strip: 4 N-tiles -> 4 independent wmma chains sharing the
// A fragment (pipelines the matrix unit; B tiles at immediate offsets).
__global__ void gemm2(const float* __restrict__ U, const float* __restrict__ H2S,
                      const float* __restrict__ C, float* __restrict__ out) {
  const int lane = threadIdx.x;
  const int half = lane >> 4;
  const int l16  = lane & 15;
  const int mt = blockIdx.x;            // 0..15  : M tile (rows of bo)
  const int n0 = blockIdx.y * 4;        // 0..1023: first of 4 N tiles (cols m)

  v8f acc[4];
  const v8f vzero = {0.f,0.f,0.f,0.f,0.f,0.f,0.f,0.f};
#pragma unroll
  for (int t = 0; t < 4; ++t) acc[t] = vzero;

  const float* aptr = U   + (long)(mt * 16 + l16) * HID + 2 * half;
  const float* bptr = H2S + ((long)n0 * 16 + l16) * HID + 2 * half;
#pragma unroll
  for (int kb = 0; kb < HID; kb += 4) {
    v2f a = *(const v2f*)(aptr + kb);
#pragma unroll
    for (int t = 0; t < 4; ++t) {
      v2f b = *(const v2f*)(bptr + t * 16 * HID + kb);   // +8 KB per tile
      acc[t] = wmma4(a, b, acc[t]);
    }
  }
  const float scale = 1.0f / (float)NX;
#pragma unroll
  for (int t = 0; t < 4; ++t)
#pragma unroll
    for (int r = 0; r < 8; ++r) {
      const int row = mt * 16 + r + 8 * half;
      out[(long)row * NX + (n0 + t) * 16 + l16] = (acc[t][r] + C[row]) * scale;
    }
}

extern "C" void kernel_launch(void* const* d_in, const int* in_sizes, int n_in,
                              void* d_out, int out_size, void* d_ws, size_t ws_size,
                              hipStream_t stream) {
  const float* x   = (const float*)d_in[0];
  const float* pw1 = (const float*)d_in[1];
  const float* pb1 = (const float*)d_in[2];
  const float* pw2 = (const float*)d_in[3];
  const float* pb2 = (const float*)d_in[4];
  const float* pw3 = (const float*)d_in[5];
  const float* pb3 = (const float*)d_in[6];
  const float* qw1 = (const float*)d_in[7];
  const float* qb1 = (const float*)d_in[8];
  const float* qw2 = (const float*)d_in[9];
  const float* qb2 = (const float*)d_in[10];
  const float* qw3 = (const float*)d_in[11];
  const float* qb3 = (const float*)d_in[12];

  float* ws  = (float*)d_ws;
  float* H2P = ws;                       // [128][16384]  phi trunk, feature-major
  float* H2S = H2P + HID * NX;           // [16384][128]  psi trunk, point-major
  float* P   = H2S + NX * HID;           // [64][256][128] GEMM1 partials
  float* T   = P + 64 * ROWS * HID;      // [256][128]
  float* X1  = T + ROWS * HID;           // [256]
  float* S   = X1 + ROWS;                // [8][32][32]
  float* Uu  = S + 8192;                 // [256][128]
  float* Cc  = Uu + ROWS * HID;          // [256]

  mlp_trunk<<<dim3(NX / 128), dim3(128), 0, stream>>>(pw1, pb1, pw2, pb2,
                                                      qw1, qb1, qw2, qb2, H2P, H2S);
  rowsum<<<dim3(ROWS), dim3(256), 0, stream>>>(x, X1);
  gemm1_partial<<<dim3(16, 64), dim3(32), 0, stream>>>(x, H2P, P);
  reduceP<<<dim3(128), dim3(256), 0, stream>>>(P, T);
  s_kernel<<<dim3(32), dim3(256), 0, stream>>>(T, pw3, pb3, X1, S);
  u_kernel<<<dim3(128), dim3(256), 0, stream>>>(S, qw3, Uu);
  c_kernel<<<dim3(1), dim3(256), 0, stream>>>(S, qb3, Cc);
  gemm2<<<dim3(16, 256), dim3(32), 0, stream>>>(Uu, H2S, Cc, (float*)d_out);
}